// DenseEnergyLoss_3573412790928
// MI455X (gfx1250) — compile-verified
//
#include <hip/hip_runtime.h>
#include <hip/hip_bf16.h>

// ---------------------------------------------------------------------------
// DenseEnergyLoss on MI455X (gfx1250), FP32 WMMA path.
//
// Problem sizes (fixed by reference): N=4, K=21, H=W=128 -> h=w=64, P=4096.
// Pair-space tiled 16x16; per tile: 2x V_WMMA_F32_16X16X4_F32 for the 5-dim
// feature dot (K padded to 8) and 6x for the 21-dim seg Gram (K padded to 24).
// ---------------------------------------------------------------------------

#define NB      4
#define KC      21
#define KPAD    24
#define FPAD    8
#define HH      64          // downsampled h
#define WW      64          // downsampled w
#define P       (HH * WW)   // 4096 pixels per batch
#define NT      (P / 16)    // 256 tiles per axis
#define SIG_RGB 15.0f
#define SIG_XY  50.0f       // SIGMA_XY * SCALE = 100 * 0.5
#define WEIGHT  1e-7f

typedef __attribute__((ext_vector_type(2))) float v2f;
typedef __attribute__((ext_vector_type(8))) float v8f;

static __device__ __forceinline__ v8f wmma4(v2f a, v2f b, v8f c) {
    // D = A(16x4,f32) x B(4x16,f32) + C(16x16,f32)
    return __builtin_amdgcn_wmma_f32_16x16x4_f32(
        /*neg_a=*/false, a, /*neg_b=*/false, b,
        /*c_mod=*/(short)0, c, /*reuse_a=*/false, /*reuse_b=*/false);
}

// ---------------------------------------------------------------------------
// Kernel 1: downsample + build padded feature / seg_r / |f|^2 / gate arrays.
//   F    [NB][P][FPAD]  features (x/50, y/50, r/15, g/15, b/15, 0,0,0)
//   S    [NB][P][KPAD]  seg_r = bilinear-down(seg) * roi, padded with zeros
//   SQ   [NB][P]        |f|^2
//   GATE [NB][P]        gate per pixel
// ---------------------------------------------------------------------------
__global__ void del_prep_kernel(const float* __restrict__ img,
                                const float* __restrict__ seg,
                                const float* __restrict__ roi,
                                const int*   __restrict__ lbl,
                                float* __restrict__ F, float* __restrict__ S,
                                float* __restrict__ SQ, float* __restrict__ GATE) {
    int tid = blockIdx.x * blockDim.x + threadIdx.x;
    if (tid >= NB * P) return;
    int n = tid / P;
    int p = tid - n * P;
    int y = p / WW;
    int x = p - y * WW;
    int Y = 2 * y, X = 2 * x;

    // nearest-downsampled image -> features
    float r = img[(((size_t)n * 3 + 0) * 128 + Y) * 128 + X];
    float g = img[(((size_t)n * 3 + 1) * 128 + Y) * 128 + X];
    float b = img[(((size_t)n * 3 + 2) * 128 + Y) * 128 + X];
    float f0 = (float)x / SIG_XY;
    float f1 = (float)y / SIG_XY;
    float f2 = r / SIG_RGB;
    float f3 = g / SIG_RGB;
    float f4 = b / SIG_RGB;

    float* Fp = F + ((size_t)n * P + p) * FPAD;
    Fp[0] = f0; Fp[1] = f1; Fp[2] = f2; Fp[3] = f3; Fp[4] = f4;
    Fp[5] = 0.f; Fp[6] = 0.f; Fp[7] = 0.f;
    SQ[(size_t)n * P + p] = f0*f0 + f1*f1 + f2*f2 + f3*f3 + f4*f4;

    float rv  = roi[((size_t)n * 128 + Y) * 128 + X];
    int   lab = lbl[((size_t)n * 128 + Y) * 128 + X];

    // bilinear down (2x2 avg) of seg, seg_r = seg*roi, track max over classes
    float smax = -3.4e38f;
    float* Sp = S + ((size_t)n * P + p) * KPAD;
    for (int k = 0; k < KC; ++k) {
        const float* sp = seg + (((size_t)n * KC + k) * 128 + Y) * 128 + X;
        float avg = 0.25f * (sp[0] + sp[1] + sp[128] + sp[129]);
        smax = fmaxf(smax, avg);
        Sp[k] = avg * rv;
    }
    for (int k = KC; k < KPAD; ++k) Sp[k] = 0.f;

    float gate = rv - smax;
    if (lab == 255) gate = 1.0f;
    GATE[(size_t)n * P + p] = fmaxf(gate, 0.f);
}

// ---------------------------------------------------------------------------
// Kernel 2: WMMA tile sweep over the pair space.
// grid = NB*NT blocks (one i-tile each), 4 waves/block; wave w handles
// j-tiles w, w+4, ... (64 each, uniform -> EXEC all-ones for WMMA).
// ---------------------------------------------------------------------------
__global__ void __launch_bounds__(128)
del_pair_kernel(const float* __restrict__ F, const float* __restrict__ S,
                const float* __restrict__ SQ, const float* __restrict__ GATE,
                float* __restrict__ partials) {
    int bid  = blockIdx.x;
    int n    = bid >> 8;          // batch
    int tI   = bid & (NT - 1);    // i-tile
    int wave = threadIdx.x >> 5;
    int lane = threadIdx.x & 31;
    int half = lane >> 4;         // selects K-pair within a 16x4 A/B chunk
    int mrow = lane & 15;         // M for A-side load, N for B-side load / C

    const float* Fb  = F    + (size_t)n * P * FPAD;
    const float* Sb  = S    + (size_t)n * P * KPAD;
    const float* SQb = SQ   + (size_t)n * P;
    const float* Gb  = GATE + (size_t)n * P;

    const int iBase = tI * 16;

    // ---- hoisted i-tile (A-side) state: invariant over the j-loop ----
    int apix = iBase + mrow;
    v2f fa0 = *(const v2f*)(Fb + (size_t)apix * FPAD + half * 2);       // K 0..3
    v2f fa1 = *(const v2f*)(Fb + (size_t)apix * FPAD + 4 + half * 2);   // K 4..7
    v2f sa[6];
#pragma unroll
    for (int c = 0; c < 6; ++c)
        sa[c] = *(const v2f*)(Sb + (size_t)apix * KPAD + c * 4 + half * 2);

    float sqi[8], gi[8];
#pragma unroll
    for (int rr = 0; rr < 8; ++rr) {
        int i = iBase + rr + 8 * half;      // C/D row held by this lane, VGPR rr
        sqi[rr] = SQb[i];
        gi[rr]  = Gb[i];
    }

    float acc = 0.f;
    for (int tJ = wave; tJ < NT; tJ += 4) {
        int jBase = tJ * 16;
        int bpix  = jBase + mrow;

        // B-side loads (L2-resident); same striping as A-side
        v2f fb0 = *(const v2f*)(Fb + (size_t)bpix * FPAD + half * 2);
        v2f fb1 = *(const v2f*)(Fb + (size_t)bpix * FPAD + 4 + half * 2);
        v2f sb[6];
#pragma unroll
        for (int c = 0; c < 6; ++c)
            sb[c] = *(const v2f*)(Sb + (size_t)bpix * KPAD + c * 4 + half * 2);
        float sqj = SQb[bpix];

        // prefetch next j-tile's seg chunk (emits global_prefetch_b8)
        __builtin_prefetch(Sb + (size_t)(bpix + 64) * KPAD + half * 2, 0, 1);

        // feature inner products: dot(f_i, f_j), K padded 5->8
        v8f dotv = {};
        dotv = wmma4(fa0, fb0, dotv);
        dotv = wmma4(fa1, fb1, dotv);

        // Gram over classes: sum_k seg_r[k,i]*seg_r[k,j], K padded 21->24
        v8f gram = {};
#pragma unroll
        for (int c = 0; c < 6; ++c)
            gram = wmma4(sa[c], sb[c], gram);

        // finish: gate[i] * exp(-0.5*d2) * gram, 8 (M) rows per lane
        float tsum = 0.f;
#pragma unroll
        for (int rr = 0; rr < 8; ++rr) {
            float d2   = fmaxf(sqi[rr] + sqj - 2.0f * dotv[rr], 0.f);
            float kern = __expf(-0.5f * d2);
            tsum = fmaf(gi[rr] * kern, gram[rr], tsum);
        }
        acc += tsum;
    }

    // wave32 reduction, then cross-wave via LDS; one partial per block
#pragma unroll
    for (int off = 16; off > 0; off >>= 1)
        acc += __shfl_down(acc, off, 32);
    __shared__ float wsum[4];
    if (lane == 0) wsum[wave] = acc;
    __syncthreads();
    if (threadIdx.x == 0)
        partials[bid] = wsum[0] + wsum[1] + wsum[2] + wsum[3];
}

// ---------------------------------------------------------------------------
// Kernel 3: deterministic final reduction (fixed summation order).
// ---------------------------------------------------------------------------
__global__ void del_final_kernel(const float* __restrict__ partials,
                                 float* __restrict__ out) {
    if (blockIdx.x == 0 && threadIdx.x == 0) {
        float t = 0.f;
        for (int i = 0; i < NB * NT; ++i) t += partials[i];
        out[0] = -(WEIGHT / (float)NB) * t;
    }
}

extern "C" void kernel_launch(void* const* d_in, const int* in_sizes, int n_in,
                              void* d_out, int out_size, void* d_ws, size_t ws_size,
                              hipStream_t stream) {
    (void)in_sizes; (void)n_in; (void)out_size; (void)ws_size;
    const float* img = (const float*)d_in[0];   // [4,3,128,128]
    const float* seg = (const float*)d_in[1];   // [4,21,128,128]
    const float* roi = (const float*)d_in[2];   // [4,128,128]
    const int*   lbl = (const int*)d_in[3];     // [4,1,128,128]
    float* out = (float*)d_out;

    // workspace layout (floats)
    float* F        = (float*)d_ws;                 // NB*P*FPAD = 131072
    float* S        = F + (size_t)NB * P * FPAD;    // NB*P*KPAD = 393216
    float* SQ       = S + (size_t)NB * P * KPAD;    // NB*P      =  16384
    float* GATE     = SQ + (size_t)NB * P;          // NB*P      =  16384
    float* partials = GATE + (size_t)NB * P;        // NB*NT     =   1024

    {
        int total = NB * P;
        int blk = 256;
        del_prep_kernel<<<(total + blk - 1) / blk, blk, 0, stream>>>(
            img, seg, roi, lbl, F, S, SQ, GATE);
    }
    del_pair_kernel<<<NB * NT, 128, 0, stream>>>(F, S, SQ, GATE, partials);
    del_final_kernel<<<1, 32, 0, stream>>>(partials, out);
}